// MT_GCN_90305982366368
// MI455X (gfx1250) — compile-verified
//
#include <hip/hip_runtime.h>
#include <hip/hip_bf16.h>

typedef __attribute__((ext_vector_type(2))) float v2f;
typedef __attribute__((ext_vector_type(8))) float v8f;

#define KC 32
#define ASTRIDE 36   // 64 x KC A-tile, padded row stride (144B, 16B aligned)
#define BSTRIDE 68   // KC x 64 B-tile, padded row stride (272B, 16B aligned)

// Truncate a generic pointer to an LDS variable down to its addrspace(3)
// byte offset (low 32 bits of the shared aperture address).
__device__ __forceinline__ unsigned lds_off(const void* p) {
    return (unsigned)(unsigned long long)(uintptr_t)p;
}

// CDNA5 async memory->LDS copies (tracked by ASYNCcnt, no VGPR staging).
__device__ __forceinline__ void async_ld_b32(unsigned lds, const float* g) {
    asm volatile("global_load_async_to_lds_b32 %0, %1, off"
                 :: "v"(lds), "v"(g) : "memory");
}
__device__ __forceinline__ void async_ld_b128(unsigned lds, const float* g) {
    asm volatile("global_load_async_to_lds_b128 %0, %1, off"
                 :: "v"(lds), "v"(g) : "memory");
}
__device__ __forceinline__ void wait_async0() {
    asm volatile("s_wait_asynccnt 0x0" ::: "memory");
}

// ---------------------------------------------------------------------------
// fp32 WMMA GEMM:  C[M,N] = A[M,K] @ B[K,N] (+ bias) (+ ReLU)
// Block: 128 threads = 4 waves. Block tile 64(M) x 64(N); wave w owns rows
// [m0+16w, +16) and all 64 cols via 4 accumulators (A-fragment reused 4x).
// Interior chunks staged with global_load_async_to_lds; edge chunks use
// guarded scalar staging. WMMAs are unconditional (EXEC all-ones).
// ---------------------------------------------------------------------------
__global__ __launch_bounds__(128)
void wmma_gemm_f32(const float* __restrict__ A,
                   const float* __restrict__ B,
                   const float* __restrict__ bias,
                   float* __restrict__ C,
                   int M, int N, int K, int relu)
{
    __shared__ __align__(16) float As[64 * ASTRIDE];
    __shared__ __align__(16) float Bs[KC * BSTRIDE];

    const int t    = threadIdx.x;   // 0..127
    const int wave = t >> 5;        // 0..3  -> M sub-tile
    const int lane = t & 31;
    const int m0   = blockIdx.y * 64;
    const int n0   = blockIdx.x * 64;

    const unsigned abase = lds_off(As);
    const unsigned bbase = lds_off(Bs);
    const bool mn_full = (m0 + 64 <= M) && (n0 + 64 <= N);

    v8f acc0 = {}, acc1 = {}, acc2 = {}, acc3 = {};

    for (int kc = 0; kc < K; kc += KC) {
        if (mn_full && (kc + KC <= K)) {
            // ---- async fast path: A via b32, B via b128 (16B-aligned) ----
            #pragma unroll
            for (int q = 0; q < 16; ++q) {          // 64x32 A tile, b32 each
                int i = t + 128 * q;
                int r = i >> 5, c = i & 31;
                async_ld_b32(abase + (unsigned)(r * ASTRIDE + c) * 4u,
                             A + (size_t)(m0 + r) * K + kc + c);
            }
            #pragma unroll
            for (int q = 0; q < 4; ++q) {           // 32x64 B tile, b128 each
                int cchunk = t + 128 * q;           // of 512 float4 chunks
                int r = cchunk >> 4, c4 = (cchunk & 15) << 2;
                async_ld_b128(bbase + (unsigned)(r * BSTRIDE + c4) * 4u,
                              B + (size_t)(kc + r) * N + n0 + c4);
            }
            wait_async0();
        } else {
            // ---- guarded scalar staging for boundary tiles ----
            for (int i = t; i < 64 * KC; i += 128) {
                int r = i >> 5, c = i & 31;
                int gr = m0 + r, gc = kc + c;
                As[r * ASTRIDE + c] =
                    (gr < M && gc < K) ? A[(size_t)gr * K + gc] : 0.0f;
            }
            for (int i = t; i < KC * 64; i += 128) {
                int r = i >> 6, c = i & 63;
                int gr = kc + r, gc = n0 + c;
                Bs[r * BSTRIDE + c] =
                    (gr < K && gc < N) ? B[(size_t)gr * N + gc] : 0.0f;
            }
        }
        __syncthreads();

        const int koff = (lane >= 16) ? 2 : 0;   // lanes 16-31 carry K+2,K+3
        const int mrow = lane & 15;              // M row in A / N col in B
        const int arow = (wave * 16 + mrow) * ASTRIDE;
        #pragma unroll
        for (int kk = 0; kk < KC; kk += 4) {
            v2f a, b0, b1, b2, b3;
            a[0]  = As[arow + kk + koff];
            a[1]  = As[arow + kk + koff + 1];
            const int br0 = (kk + koff) * BSTRIDE + mrow;
            const int br1 = br0 + BSTRIDE;
            b0[0] = Bs[br0];      b0[1] = Bs[br1];
            b1[0] = Bs[br0 + 16]; b1[1] = Bs[br1 + 16];
            b2[0] = Bs[br0 + 32]; b2[1] = Bs[br1 + 32];
            b3[0] = Bs[br0 + 48]; b3[1] = Bs[br1 + 48];
            acc0 = __builtin_amdgcn_wmma_f32_16x16x4_f32(
                false, a, false, b0, (short)0, acc0, false, false);
            acc1 = __builtin_amdgcn_wmma_f32_16x16x4_f32(
                false, a, false, b1, (short)0, acc1, false, false);
            acc2 = __builtin_amdgcn_wmma_f32_16x16x4_f32(
                false, a, false, b2, (short)0, acc2, false, false);
            acc3 = __builtin_amdgcn_wmma_f32_16x16x4_f32(
                false, a, false, b3, (short)0, acc3, false, false);
        }
        __syncthreads();
    }

    // store: C/D layout -> VGPR v: lanes 0-15 row v, lanes 16-31 row v+8
    const int mbase = m0 + wave * 16 + ((lane >= 16) ? 8 : 0);
    const int cbase = n0 + (lane & 15);
    v8f accs[4] = {acc0, acc1, acc2, acc3};
    #pragma unroll
    for (int j = 0; j < 4; ++j) {
        const int col = cbase + j * 16;
        if (col < N) {
            const float bv = bias ? bias[col] : 0.0f;
            #pragma unroll
            for (int v = 0; v < 8; ++v) {
                int row = mbase + v;
                if (row < M) {
                    float val = accs[j][v] + bv;
                    if (relu) val = fmaxf(val, 0.0f);
                    C[(size_t)row * N + col] = val;
                }
            }
        }
    }
}

// ---------------------------------------------------------------------------
// Elementwise / scatter helpers
// ---------------------------------------------------------------------------
__global__ void fill_kernel(float* __restrict__ p, float v, long n)
{
    long i = (long)blockIdx.x * blockDim.x + threadIdx.x;
    if (i < n) p[i] = v;
}

__global__ void degree_kernel(const int* __restrict__ dst,
                              float* __restrict__ deg, int E)
{
    int e = blockIdx.x * blockDim.x + threadIdx.x;
    if (e < E) atomicAdd(&deg[dst[e]], 1.0f);
}

__global__ void rsqrt_kernel(float* __restrict__ d, int n)
{
    int i = blockIdx.x * blockDim.x + threadIdx.x;
    if (i < n) d[i] = rsqrtf(d[i]);
}

// One block per edge, one thread per feature: agg[dst] += norm * h[src]
__global__ void edge_aggregate(const int* __restrict__ src,
                               const int* __restrict__ dst,
                               const float* __restrict__ dinv,
                               const float* __restrict__ h,
                               float* __restrict__ agg,
                               int E, int fshift)
{
    int e = blockIdx.x;
    if (e >= E) return;
    int s = src[e], d = dst[e];
    float norm = dinv[s] * dinv[d];
    int f = threadIdx.x;
    atomicAdd(&agg[((size_t)d << fshift) + f],
              norm * h[((size_t)s << fshift) + f]);
}

// agg += h * dinv^2 + bias  (self-loop contribution), optional ReLU
__global__ void self_loop_bias(const float* __restrict__ h,
                               const float* __restrict__ dinv,
                               const float* __restrict__ bias,
                               float* __restrict__ agg,
                               long total, int fshift, int relu)
{
    long idx = (long)blockIdx.x * blockDim.x + threadIdx.x;
    if (idx >= total) return;
    int n = (int)(idx >> fshift);
    int f = (int)(idx & ((1 << fshift) - 1));
    float di = dinv[n];
    float v = agg[idx] + h[idx] * di * di + bias[f];
    if (relu) v = fmaxf(v, 0.0f);
    agg[idx] = v;
}

// global sum pool: g[batch[n]] += a[n]   (F = 64)
__global__ void segment_pool(const float* __restrict__ a,
                             const int* __restrict__ batch,
                             float* __restrict__ g, long total)
{
    long idx = (long)blockIdx.x * blockDim.x + threadIdx.x;
    if (idx >= total) return;
    int n = (int)(idx >> 6);
    int f = (int)(idx & 63);
    atomicAdd(&g[((size_t)batch[n] << 6) + f], a[idx]);
}

// z0[G,75] = concat(g[G,64], solvent[G,11])
__global__ void concat_kernel(const float* __restrict__ g,
                              const float* __restrict__ solv,
                              float* __restrict__ z0, int G)
{
    long idx = (long)blockIdx.x * blockDim.x + threadIdx.x;
    long total = (long)G * 75;
    if (idx >= total) return;
    int r = (int)(idx / 75);
    int c = (int)(idx % 75);
    z0[idx] = (c < 64) ? g[(size_t)r * 64 + c] : solv[(size_t)r * 11 + (c - 64)];
}

// 6 heads: out[g,h] = sum_k relu(z3[g]·Wh1[h,:,k] + bh1[h,k]) * Wh2[h,k] + bh2[h]
// Block = 192 threads = 6 wave32s; wave h handles head h, lane k = unit k.
__global__ void heads_kernel(const float* __restrict__ z3,
                             const float* __restrict__ Wh1,
                             const float* __restrict__ bh1,
                             const float* __restrict__ Wh2,
                             const float* __restrict__ bh2,
                             float* __restrict__ out, int G)
{
    __shared__ float zs[128];
    int gidx = blockIdx.x;
    if (gidx >= G) return;
    int t = threadIdx.x;           // 0..191
    if (t < 128) zs[t] = z3[(size_t)gidx * 128 + t];
    __syncthreads();

    int h = t >> 5;                // head = wave id (0..5)
    int k = t & 31;                // hidden unit
    const float* w = Wh1 + (size_t)h * 128 * 32 + k;
    float dot = 0.0f;
    #pragma unroll 8
    for (int f = 0; f < 128; ++f) dot += zs[f] * w[(size_t)f * 32];
    float a = fmaxf(dot + bh1[h * 32 + k], 0.0f);
    float p = a * Wh2[h * 32 + k];
    #pragma unroll
    for (int off = 16; off > 0; off >>= 1) p += __shfl_down(p, off, 32);
    if (k == 0) out[(size_t)gidx * 6 + h] = p + bh2[h];
}

// ---------------------------------------------------------------------------
// Orchestration
// ---------------------------------------------------------------------------
static inline int ceil_div(long a, long b) { return (int)((a + b - 1) / b); }

extern "C" void kernel_launch(void* const* d_in, const int* in_sizes, int n_in,
                              void* d_out, int out_size, void* d_ws, size_t ws_size,
                              hipStream_t stream)
{
    const float* x    = (const float*)d_in[0];
    const int*   esrc = (const int*)d_in[1];
    const int*   edst = (const int*)d_in[2];
    const int*   batch = (const int*)d_in[3];
    const float* solv = (const float*)d_in[4];
    const float* Wc1 = (const float*)d_in[5];  const float* bc1 = (const float*)d_in[6];
    const float* Wc2 = (const float*)d_in[7];  const float* bc2 = (const float*)d_in[8];
    const float* W1  = (const float*)d_in[9];  const float* b1  = (const float*)d_in[10];
    const float* W2  = (const float*)d_in[11]; const float* b2  = (const float*)d_in[12];
    const float* W3  = (const float*)d_in[13]; const float* b3  = (const float*)d_in[14];
    const float* Wh1 = (const float*)d_in[15]; const float* bh1 = (const float*)d_in[16];
    const float* Wh2 = (const float*)d_in[17]; const float* bh2 = (const float*)d_in[18];

    const int N = in_sizes[3];           // nodes (batch_index length)
    const int E = in_sizes[1];           // edges
    const int G = out_size / 6;          // graphs

    float* ws = (float*)d_ws;
    size_t o = 0;
    float* dinv = ws + o; o += (size_t)N;
    float* h1   = ws + o; o += (size_t)N * 128;
    float* a1   = ws + o; o += (size_t)N * 128;
    float* h2   = ws + o; o += (size_t)N * 64;
    float* a2   = ws + o; o += (size_t)N * 64;
    float* gp   = ws + o; o += (size_t)G * 64;
    float* z0   = ws + o; o += (size_t)G * 75;
    float* z1 = h1;   // reuse: h1 dead after conv2 GEMM
    float* z2 = h2;   // reuse: h2 dead after pooling
    float* z3 = a2;   // reuse: a2 dead after pooling
    float* outp = (float*)d_out;

    // init accumulators
    fill_kernel<<<ceil_div(N, 256), 256, 0, stream>>>(dinv, 1.0f, N);
    fill_kernel<<<ceil_div((long)N * 128, 256), 256, 0, stream>>>(a1, 0.0f, (long)N * 128);
    fill_kernel<<<ceil_div((long)N * 64, 256), 256, 0, stream>>>(a2, 0.0f, (long)N * 64);
    fill_kernel<<<ceil_div((long)G * 64, 256), 256, 0, stream>>>(gp, 0.0f, (long)G * 64);

    // symmetric norm: dinv = rsqrt(deg + 1)
    degree_kernel<<<ceil_div(E, 256), 256, 0, stream>>>(edst, dinv, E);
    rsqrt_kernel<<<ceil_div(N, 256), 256, 0, stream>>>(dinv, N);

    // ---- GCN layer 1: h1 = x @ Wc1 ; a1 = relu(scatter + selfloop + bc1)
    wmma_gemm_f32<<<dim3(2, ceil_div(N, 64)), 128, 0, stream>>>(
        x, Wc1, nullptr, h1, N, 128, 131, 0);
    edge_aggregate<<<E, 128, 0, stream>>>(esrc, edst, dinv, h1, a1, E, 7);
    self_loop_bias<<<ceil_div((long)N * 128, 256), 256, 0, stream>>>(
        h1, dinv, bc1, a1, (long)N * 128, 7, 1);

    // ---- GCN layer 2: h2 = a1 @ Wc2 ; a2 = scatter + selfloop + bc2
    wmma_gemm_f32<<<dim3(1, ceil_div(N, 64)), 128, 0, stream>>>(
        a1, Wc2, nullptr, h2, N, 64, 128, 0);
    edge_aggregate<<<E, 64, 0, stream>>>(esrc, edst, dinv, h2, a2, E, 6);
    self_loop_bias<<<ceil_div((long)N * 64, 256), 256, 0, stream>>>(
        h2, dinv, bc2, a2, (long)N * 64, 6, 0);

    // ---- pool + concat
    segment_pool<<<ceil_div((long)N * 64, 256), 256, 0, stream>>>(
        a2, batch, gp, (long)N * 64);
    concat_kernel<<<ceil_div((long)G * 75, 256), 256, 0, stream>>>(gp, solv, z0, G);

    // ---- MLP (all fp32 WMMA, fused bias+ReLU)
    wmma_gemm_f32<<<dim3(32, ceil_div(G, 64)), 128, 0, stream>>>(
        z0, W1, b1, z1, G, 2048, 75, 1);
    wmma_gemm_f32<<<dim3(16, ceil_div(G, 64)), 128, 0, stream>>>(
        z1, W2, b2, z2, G, 1024, 2048, 1);
    wmma_gemm_f32<<<dim3(2, ceil_div(G, 64)), 128, 0, stream>>>(
        z2, W3, b3, z3, G, 128, 1024, 1);

    // ---- property heads
    heads_kernel<<<G, 192, 0, stream>>>(z3, Wh1, bh1, Wh2, bh2, outp, G);
}